// TKGCN_86526411145584
// MI455X (gfx1250) — compile-verified
//
#include <hip/hip_runtime.h>

typedef __attribute__((ext_vector_type(2))) float v2f;
typedef __attribute__((ext_vector_type(8))) float v8f;

#define N_NODES 16384
#define IN_F    128
#define OUT_F   128
#define TOP_K   32

// ---------------------------------------------------------------------------
// Kernel 1: h = x @ W^T + b  via V_WMMA_F32_16X16X4_F32 (fp32 tensor path)
// ---------------------------------------------------------------------------
__global__ __launch_bounds__(256) void tkgcn_gemm_wmma(
    const float* __restrict__ x, const float* __restrict__ W,
    const float* __restrict__ b, float* __restrict__ h) {
  const int lane = threadIdx.x & 31;
  const int wave = threadIdx.x >> 5;
  const int hi   = lane >> 4;
  const int m    = lane & 15;
  const int r0   = blockIdx.x * 16;
  const int o0   = wave * 16;

  const float* arow = x + (size_t)(r0 + m) * IN_F + 2 * hi;
  const float* brow = W + (size_t)(o0 + m) * IN_F + 2 * hi;

  v8f acc = {0.f, 0.f, 0.f, 0.f, 0.f, 0.f, 0.f, 0.f};
#pragma unroll
  for (int k0 = 0; k0 < IN_F; k0 += 4) {
    v2f a  = *(const v2f*)(arow + k0);
    v2f bb = *(const v2f*)(brow + k0);
    acc = __builtin_amdgcn_wmma_f32_16x16x4_f32(
        false, a, false, bb, (short)0, acc, false, false);
  }
  const float bias = b[o0 + m];
#pragma unroll
  for (int r = 0; r < 8; ++r)
    h[(size_t)(r0 + r + 8 * hi) * OUT_F + o0 + m] = acc[r] + bias;
}

// ---------------------------------------------------------------------------
// Kernel 2: exact top-32 per row + gather/max.
// ---------------------------------------------------------------------------
#if __has_builtin(__builtin_amdgcn_global_load_async_to_lds_b128) && \
    __has_builtin(__builtin_amdgcn_s_wait_asynccnt)
#define USE_ASYNC_LDS 1
// Builtin parameter type per hipcc diagnostic:
//   "__attribute__((__vector_size__(4 * sizeof(int)))) int __device__ *"
// i.e. pointer-to-int-vector(16B) in address_space(1) (and (3) for LDS).
typedef int v4i_vs __attribute__((__vector_size__(16)));
#define AS1P(p) ((__attribute__((address_space(1))) v4i_vs*)(unsigned long long)(p))
#define AS3P(p) ((__attribute__((address_space(3))) v4i_vs*)(unsigned long long)(p))
#else
#define USE_ASYNC_LDS 0
#endif

__device__ __forceinline__ unsigned fkey(float f) {
  unsigned u = __float_as_uint(f);
  return u ^ ((unsigned)((int)u >> 31) | 0x80000000u);
}
__device__ __forceinline__ unsigned long long packkey(float f, int idx) {
  // high 32: monotonic value bits; low 32: ~idx (unsigned max => lowest index
  // wins on value ties, matching jax.lax.top_k).
  return ((unsigned long long)fkey(f) << 32) | (unsigned)~idx;
}
__device__ __forceinline__ unsigned long long ullmax(unsigned long long a,
                                                     unsigned long long b) {
  return a < b ? b : a;
}
__device__ __forceinline__ unsigned long long shflxor64(unsigned long long v,
                                                        int mask) {
  unsigned lo = __shfl_xor((unsigned)v, mask, 32);
  unsigned hi = __shfl_xor((unsigned)(v >> 32), mask, 32);
  return ((unsigned long long)hi << 32) | lo;
}
// XOR bank swizzle on float4 slots: re-scan slots {c + 256j} map to 16
// distinct bank groups; coalesced copy stays conflict-free.
__device__ __forceinline__ int swz4(int i4) { return i4 ^ ((i4 >> 8) & 15); }

__global__ __launch_bounds__(256) void tkgcn_topk_gather(
    const float* __restrict__ adj, const float* __restrict__ h,
    float* __restrict__ out) {
  __shared__ float              row[N_NODES];   // 64 KB, swizzled f4 layout
  __shared__ unsigned long long ckey[256];      // chunk-key handoff to wave 0
  __shared__ int                sel[TOP_K];
  __shared__ float              part[256];

  const int t = threadIdx.x;
  const int n = blockIdx.x;
  const float4* src4 = (const float4*)(adj + (size_t)n * N_NODES);
  float4* row4 = (float4*)row;

  // ---- Stage row into LDS; chunk t := the 64 elements thread t streams. ----
#if USE_ASYNC_LDS
#pragma unroll
  for (int j = 0; j < 16; ++j) {
    const int i4 = t + 256 * j;
    __builtin_amdgcn_global_load_async_to_lds_b128(
        AS1P(src4 + i4), AS3P(row4 + swz4(i4)), 0, 0);
  }
  __builtin_amdgcn_s_wait_asynccnt(0);
  __syncthreads();
  {
    unsigned long long best = 0;
#pragma unroll 4
    for (int j = 0; j < 16; ++j) {
      const int i4 = t + 256 * j;
      const float4 v = row4[swz4(i4)];           // conflict-free sweep
      best = ullmax(best, packkey(v.x, 4 * i4 + 0));
      best = ullmax(best, packkey(v.y, 4 * i4 + 1));
      best = ullmax(best, packkey(v.z, 4 * i4 + 2));
      best = ullmax(best, packkey(v.w, 4 * i4 + 3));
    }
    ckey[t] = best;
  }
#else
  {
    unsigned long long best = 0;
#pragma unroll 4
    for (int j = 0; j < 16; ++j) {
      const int i4 = t + 256 * j;
      const float4 v = src4[i4];                 // coalesced b128 stream
      row4[swz4(i4)] = v;
      best = ullmax(best, packkey(v.x, 4 * i4 + 0));
      best = ullmax(best, packkey(v.y, 4 * i4 + 1));
      best = ullmax(best, packkey(v.z, 4 * i4 + 2));
      best = ullmax(best, packkey(v.w, 4 * i4 + 3));
    }
    ckey[t] = best;
  }
#endif
  __syncthreads();

  // ---- Wave 0: 32 selection rounds, chunk keys held in registers. ----
  if (t < 32) {
    unsigned long long k0 = ckey[t +   0], k1 = ckey[t +  32];
    unsigned long long k2 = ckey[t +  64], k3 = ckey[t +  96];
    unsigned long long k4 = ckey[t + 128], k5 = ckey[t + 160];
    unsigned long long k6 = ckey[t + 192], k7 = ckey[t + 224];

    for (int round = 0; round < TOP_K; ++round) {
      unsigned long long best =
          ullmax(ullmax(ullmax(k0, k1), ullmax(k2, k3)),
                 ullmax(ullmax(k4, k5), ullmax(k6, k7)));
#pragma unroll
      for (int off = 16; off; off >>= 1)
        best = ullmax(best, shflxor64(best, off));
      const int idx = (int)(~(unsigned)best) & (N_NODES - 1);

      if (t == 0) {
        sel[round] = idx;
        const int i4 = idx >> 2;
        row[(swz4(i4) << 2) | (idx & 3)] = -__builtin_inff();
      }
      __builtin_amdgcn_wave_barrier();  // keep DS store ahead of re-scan

      // Re-scan chunk c (64 elems): lane t covers f4 j=t>>1, elems 2*(t&1)+{0,1}.
      const int c  = (idx >> 2) & 255;
      const int j  = t >> 1;
      const int e0 = 2 * (t & 1);
      const int i4 = c + 256 * j;
      const float* f4p = &row[swz4(i4) << 2];
      unsigned long long nb = ullmax(packkey(f4p[e0],     4 * i4 + e0),
                                     packkey(f4p[e0 + 1], 4 * i4 + e0 + 1));
#pragma unroll
      for (int off = 16; off; off >>= 1)
        nb = ullmax(nb, shflxor64(nb, off));

      if ((c & 31) == t) {                // uniform owner update, no spills
        switch (c >> 5) {
          case 0: k0 = nb; break; case 1: k1 = nb; break;
          case 2: k2 = nb; break; case 3: k3 = nb; break;
          case 4: k4 = nb; break; case 5: k5 = nb; break;
          case 6: k6 = nb; break; default: k7 = nb; break;
        }
      }
    }
  }
  __syncthreads();

  // ---- Gather selected h rows (L2-resident) and max-pool. ----
  {
    const int f  = t & 127;
    const int j0 = (t >> 7) * 16;
    float m = -__builtin_inff();
#pragma unroll 4
    for (int j = 0; j < 16; ++j)
      m = fmaxf(m, h[(size_t)sel[j0 + j] * OUT_F + f]);
    part[t] = m;
  }
  __syncthreads();
  if (t < 128)
    out[(size_t)n * OUT_F + t] = fmaxf(part[t], part[t + 128]);
}

// ---------------------------------------------------------------------------
extern "C" void kernel_launch(void* const* d_in, const int* in_sizes, int n_in,
                              void* d_out, int out_size, void* d_ws,
                              size_t ws_size, hipStream_t stream) {
  const float* x   = (const float*)d_in[0];   // [16384, 128]
  const float* adj = (const float*)d_in[1];   // [16384, 16384]
  const float* W   = (const float*)d_in[2];   // [128, 128]
  const float* b   = (const float*)d_in[3];   // [128]
  float* out = (float*)d_out;                 // [16384, 128]
  float* h   = (float*)d_ws;                  // 16384*128 fp32 = 8 MB scratch

  tkgcn_gemm_wmma<<<dim3(N_NODES / 16), dim3(256), 0, stream>>>(x, W, b, h);
  tkgcn_topk_gather<<<dim3(N_NODES), dim3(256), 0, stream>>>(adj, h, out);
}